// EnhancedHeteroGNN_51376398795359
// MI455X (gfx1250) — compile-verified
//
#include <hip/hip_runtime.h>
#include <hip/hip_bf16.h>
#include <math.h>

#define N_NODES 50000
#define E_EDGES 1000000
#define R_RSU   64
#define FDIM    9
#define EDIM    8
#define HHEADS  4
#define CH0     16
#define HID     64
#define NACT    10
#define KPAD    32   // IN0 = 25 padded to 32 for WMMA K-steps of 4

typedef float v2f __attribute__((ext_vector_type(2)));
typedef float v8f __attribute__((ext_vector_type(8)));

// ---- misc scalar block layout (floats) ----
#define MISC_GATES  0   // 3
#define MISC_EW     4   // 3
#define MISC_MSUM   8   // 8  (raw sums of edge_attr over E)
#define MISC_WAE0   16  // 8*4  [d*4+h], gate folded
#define MISC_WAE1   48  // 8
#define MISC_ASELF0 56  // 4
#define MISC_ASELF1 60  // 1

// ---- workspace offsets (floats) ----
constexpr size_t OFF_XPAD = 0;
constexpr size_t OFF_HLIN = OFF_XPAD + (size_t)N_NODES * KPAD;
constexpr size_t OFF_OAGG = OFF_HLIN + (size_t)N_NODES * HID;
constexpr size_t OFF_XC   = OFF_OAGG + (size_t)N_NODES * HID;
constexpr size_t OFF_AS   = OFF_XC   + (size_t)N_NODES * HID;
constexpr size_t OFF_AD   = OFF_AS   + (size_t)N_NODES * HHEADS;
constexpr size_t OFF_DEN  = OFF_AD   + (size_t)N_NODES * HHEADS;
constexpr size_t OFF_EX   = OFF_DEN  + (size_t)N_NODES * HHEADS;
constexpr size_t OFF_SC   = OFF_EX   + (size_t)(E_EDGES + N_NODES) * HHEADS;
constexpr size_t OFF_MISC = OFF_SC   + (size_t)N_NODES;
constexpr size_t OFF_W0P  = OFF_MISC + 64;
constexpr size_t OFF_POOL = OFF_W0P  + (size_t)KPAD * HID;
constexpr size_t OFF_Z    = OFF_POOL + (size_t)R_RSU * HID;
constexpr size_t OFF_Z1   = OFF_Z    + (size_t)R_RSU * 2 * HID;
constexpr size_t OFF_Z2   = OFF_Z1   + (size_t)R_RSU * HID;

// ---------------------------------------------------------------------------
__global__ void zero_kernel(float* __restrict__ p, long long n) {
  long long t = (long long)blockIdx.x * blockDim.x + threadIdx.x;
  if (t < n) p[t] = 0.f;
}

__global__ void scalars_kernel(const float* __restrict__ eg,
                               const float* __restrict__ ea,
                               float* __restrict__ misc) {
  if (threadIdx.x == 0 && blockIdx.x == 0) {
    float m = fmaxf(ea[0], fmaxf(ea[1], ea[2]));
    float e0 = expf(ea[0] - m), e1 = expf(ea[1] - m), e2 = expf(ea[2] - m);
    float s = e0 + e1 + e2;
    misc[MISC_EW + 0] = e0 / s;
    misc[MISC_EW + 1] = e1 / s;
    misc[MISC_EW + 2] = e2 / s;
    for (int i = 0; i < 3; ++i)
      misc[MISC_GATES + i] = 1.f / (1.f + expf(-eg[i]));
  }
}

__global__ void build_x_kernel(const float* __restrict__ nf,
                               const int* __restrict__ nt,
                               const float* __restrict__ temb,
                               float* __restrict__ xp) {
  long long t = (long long)blockIdx.x * blockDim.x + threadIdx.x;
  if (t >= (long long)N_NODES * KPAD) return;
  int n = (int)(t / KPAD), j = (int)(t % KPAD);
  float v = 0.f;
  if (j < FDIM)            v = nf[(size_t)n * FDIM + j];
  else if (j < FDIM + 16)  v = temb[nt[n] * 16 + (j - FDIM)];
  xp[t] = v;
}

__global__ void reduce_mean_kernel(const float* __restrict__ attr,
                                   float* __restrict__ msum, int E) {
  __shared__ float sm[EDIM];
  int t = threadIdx.x;
  if (t < EDIM) sm[t] = 0.f;
  __syncthreads();
  int e = blockIdx.x * blockDim.x + t;
  if (e < E) {
    const float* a = attr + (size_t)e * EDIM;
    for (int dd = 0; dd < EDIM; ++dd) atomicAdd(&sm[dd], a[dd]);
  }
  __syncthreads();
  if (t < EDIM) atomicAdd(&msum[t], sm[t]);
}

__global__ void prep_type_kernel(const float* __restrict__ We0,
                                 const float* __restrict__ Ae0,
                                 const float* __restrict__ We1,
                                 const float* __restrict__ Ae1,
                                 float* __restrict__ misc, int i, float invE) {
  if (threadIdx.x != 0 || blockIdx.x != 0) return;
  float gate = misc[MISC_GATES + i];
  float mean[EDIM];
  for (int dd = 0; dd < EDIM; ++dd) mean[dd] = misc[MISC_MSUM + dd] * invE;
  // layer-0: Wae0[d][h] = gate * sum_c We0[i,d,h*16+c] * Ae0[i,h,c]
  for (int dd = 0; dd < EDIM; ++dd)
    for (int h = 0; h < HHEADS; ++h) {
      float w = 0.f;
      for (int c = 0; c < CH0; ++c)
        w += We0[((size_t)i * EDIM + dd) * HID + h * CH0 + c] *
             Ae0[(size_t)i * HID + h * CH0 + c];
      misc[MISC_WAE0 + dd * HHEADS + h] = gate * w;
    }
  // layer-1: Wae1[d] = gate * sum_c We1[i,d,c] * Ae1[i,0,c]
  for (int dd = 0; dd < EDIM; ++dd) {
    float w = 0.f;
    for (int c = 0; c < HID; ++c)
      w += We1[((size_t)i * EDIM + dd) * HID + c] * Ae1[(size_t)i * HID + c];
    misc[MISC_WAE1 + dd] = gate * w;
  }
  // self-loop edge logits (ea_self = gate*mean_attr; gate already in Wae)
  for (int h = 0; h < HHEADS; ++h) {
    float v = 0.f;
    for (int dd = 0; dd < EDIM; ++dd)
      v += mean[dd] * misc[MISC_WAE0 + dd * HHEADS + h];
    misc[MISC_ASELF0 + h] = v;
  }
  float v1 = 0.f;
  for (int dd = 0; dd < EDIM; ++dd) v1 += mean[dd] * misc[MISC_WAE1 + dd];
  misc[MISC_ASELF1] = v1;
}

__global__ void padw0_kernel(const float* __restrict__ W0,
                             float* __restrict__ W0p, int i) {
  int t = blockIdx.x * blockDim.x + threadIdx.x;
  if (t >= KPAD * HID) return;
  int k = t / HID, j = t % HID;
  W0p[t] = (k < FDIM + 16) ? W0[((size_t)i * (FDIM + 16) + k) * HID + j] : 0.f;
}

// ---- f32 WMMA GEMM: C(MxN) = A(MxK) @ B(KxN); one wave per 16x16 tile ----
__global__ __launch_bounds__(32) void wmma_gemm_kernel(
    const float* __restrict__ A, const float* __restrict__ B,
    float* __restrict__ C, int K, int Nn) {
  int mt = blockIdx.x, ntile = blockIdx.y;
  int lane = threadIdx.x & 31;
  int half = lane >> 4;        // 0: K pair {0,1}, 1: K pair {2,3}
  int l16  = lane & 15;
  size_t arow = (size_t)(mt * 16 + l16) * (size_t)K;
  int col = ntile * 16 + l16;
  v8f acc = {};
  for (int k0 = 0; k0 < K; k0 += 4) {
    v2f a, b;
    a.x = A[arow + k0 + 2 * half + 0];
    a.y = A[arow + k0 + 2 * half + 1];
    b.x = B[(size_t)(k0 + 2 * half + 0) * Nn + col];
    b.y = B[(size_t)(k0 + 2 * half + 1) * Nn + col];
    acc = __builtin_amdgcn_wmma_f32_16x16x4_f32(
        /*neg_a=*/false, a, /*neg_b=*/false, b,
        /*c_mod=*/(short)0, acc, /*reuse_a=*/false, /*reuse_b=*/false);
  }
  int base = mt * 16 + 8 * half;   // VGPR r -> row r (lanes 0-15) / r+8 (16-31)
#pragma unroll
  for (int r = 0; r < 8; ++r)
    C[(size_t)(base + r) * Nn + col] = acc[r];
}

__global__ void alpha0_kernel(const float* __restrict__ h,
                              const float* __restrict__ As0,
                              const float* __restrict__ Ad0,
                              float* __restrict__ as_, float* __restrict__ ad_,
                              int i) {
  int n = blockIdx.x * blockDim.x + threadIdx.x;
  if (n >= N_NODES) return;
  const float* hn = h + (size_t)n * HID;
  for (int hh = 0; hh < HHEADS; ++hh) {
    float s = 0.f, d = 0.f;
    for (int c = 0; c < CH0; ++c) {
      float v = hn[hh * CH0 + c];
      s += v * As0[(size_t)i * HID + hh * CH0 + c];
      d += v * Ad0[(size_t)i * HID + hh * CH0 + c];
    }
    as_[n * HHEADS + hh] = s;
    ad_[n * HHEADS + hh] = d;
  }
}

__global__ void alpha1_kernel(const float* __restrict__ h,
                              const float* __restrict__ As1,
                              const float* __restrict__ Ad1,
                              float* __restrict__ as_, float* __restrict__ ad_,
                              int i) {
  int n = blockIdx.x * blockDim.x + threadIdx.x;
  if (n >= N_NODES) return;
  const float* hn = h + (size_t)n * HID;
  float s = 0.f, d = 0.f;
  for (int c = 0; c < HID; ++c) {
    float v = hn[c];
    s += v * As1[(size_t)i * HID + c];
    d += v * Ad1[(size_t)i * HID + c];
  }
  as_[n] = s;
  ad_[n] = d;
}

template <int HH>
__global__ void edgeA_kernel(const int* __restrict__ ei,
                             const float* __restrict__ attr,
                             const float* __restrict__ misc,
                             const float* __restrict__ as_,
                             const float* __restrict__ ad_,
                             float* __restrict__ ex, float* __restrict__ den,
                             int i) {
  int e = blockIdx.x * blockDim.x + threadIdx.x;
  const int etot = E_EDGES + N_NODES;
  if (e >= etot) return;
  int s, d;
  float ae[HH];
  if (e < E_EDGES) {
    s = ei[(size_t)i * 2 * E_EDGES + e];
    d = ei[(size_t)i * 2 * E_EDGES + E_EDGES + e];
    const float* at = attr + ((size_t)i * E_EDGES + e) * EDIM;
    for (int hh = 0; hh < HH; ++hh) {
      float v = 0.f;
      for (int dd = 0; dd < EDIM; ++dd)
        v += at[dd] * ((HH == 4) ? misc[MISC_WAE0 + dd * 4 + hh]
                                 : misc[MISC_WAE1 + dd]);
      ae[hh] = v;
    }
  } else {
    s = d = e - E_EDGES;
    for (int hh = 0; hh < HH; ++hh)
      ae[hh] = (HH == 4) ? misc[MISC_ASELF0 + hh] : misc[MISC_ASELF1];
  }
  for (int hh = 0; hh < HH; ++hh) {
    float al = as_[s * HH + hh] + ad_[d * HH + hh] + ae[hh];
    al = (al >= 0.f) ? al : 0.2f * al;       // leaky relu
    float ev = expf(al);                     // logits are O(1); max-shift is a no-op
    ex[(size_t)e * HH + hh] = ev;
    atomicAdd(&den[d * HH + hh], ev);
  }
}

template <int HH>
__global__ void edgeB_kernel(const int* __restrict__ ei,
                             const float* __restrict__ h,
                             const float* __restrict__ ex,
                             const float* __restrict__ den,
                             float* __restrict__ oagg, int i) {
  long long t = (long long)blockIdx.x * blockDim.x + threadIdx.x;
  const long long tot = (long long)(E_EDGES + N_NODES) * HID;
  if (t >= tot) return;
  int c = (int)(t & 63);
  int e = (int)(t >> 6);
  int s, d;
  if (e < E_EDGES) {
    s = ei[(size_t)i * 2 * E_EDGES + e];
    d = ei[(size_t)i * 2 * E_EDGES + E_EDGES + e];
  } else {
    s = d = e - E_EDGES;
  }
  int hh = (HH == 4) ? (c >> 4) : 0;
  float w = ex[(size_t)e * HH + hh] / den[d * HH + hh];
  atomicAdd(&oagg[(size_t)d * HID + c], h[(size_t)s * HID + c] * w);
}

__global__ void elu_bias_kernel(float* __restrict__ o,
                                const float* __restrict__ b0, int i) {
  long long t = (long long)blockIdx.x * blockDim.x + threadIdx.x;
  if (t >= (long long)N_NODES * HID) return;
  int c = (int)(t & 63);
  float v = o[t] + b0[(size_t)i * HID + c];
  o[t] = (v > 0.f) ? v : (expf(v) - 1.f);
}

__global__ void accum_kernel(float* __restrict__ xc,
                             const float* __restrict__ o,
                             const float* __restrict__ b1,
                             const float* __restrict__ misc, int i) {
  long long t = (long long)blockIdx.x * blockDim.x + threadIdx.x;
  if (t >= (long long)N_NODES * HID) return;
  int c = (int)(t & 63);
  xc[t] += misc[MISC_EW + i] * (o[t] + b1[(size_t)i * HID + c]);
}

__global__ void scores_kernel(const float* __restrict__ xc,
                              const float* __restrict__ pw,
                              const float* __restrict__ pb,
                              float* __restrict__ sc) {
  int j = blockIdx.x * blockDim.x + threadIdx.x;
  int n = R_RSU + j;
  if (n >= N_NODES) return;
  float s = pb[0];
  const float* xn = xc + (size_t)n * HID;
  for (int c = 0; c < HID; ++c) s += xn[c] * pw[c];
  sc[n] = s;
}

__global__ __launch_bounds__(256) void pool_kernel(
    const float* __restrict__ xc, const float* __restrict__ sc,
    const int* __restrict__ mask, float* __restrict__ pooled) {
  __shared__ float accs[4][HID];
  __shared__ float dens[4];
  int r = blockIdx.x;
  int t = threadIdx.x;
  int c = t & 63, g = t >> 6;
  const int M = N_NODES - R_RSU;
  const int* mr = mask + (size_t)r * M;
  float acc = 0.f, dl = 0.f;
  for (int j = g; j < M; j += 4) {
    if (mr[j] > 0) {
      float w = expf(sc[R_RSU + j]);
      acc += w * xc[(size_t)(R_RSU + j) * HID + c];
      if (c == 0) dl += w;
    }
  }
  accs[g][c] = acc;
  if (c == 0) dens[g] = dl;
  __syncthreads();
  if (t < HID) {
    float s = accs[0][t] + accs[1][t] + accs[2][t] + accs[3][t];
    float dsum = dens[0] + dens[1] + dens[2] + dens[3];
    pooled[(size_t)r * HID + t] = (dsum > 0.f) ? s / dsum : 0.f;
  }
}

__global__ void buildz_kernel(const float* __restrict__ xc,
                              const float* __restrict__ pooled,
                              float* __restrict__ z) {
  int t = blockIdx.x * blockDim.x + threadIdx.x;
  if (t >= R_RSU * 2 * HID) return;
  int r = t / (2 * HID), j = t % (2 * HID);
  z[t] = (j < HID) ? xc[(size_t)r * HID + j]
                   : pooled[(size_t)r * HID + (j - HID)];
}

__global__ void small_mm_kernel(const float* __restrict__ A,
                                const float* __restrict__ W,
                                const float* __restrict__ b,
                                float* __restrict__ out,
                                int M, int K, int Nn, int relu) {
  int t = blockIdx.x * blockDim.x + threadIdx.x;
  if (t >= M * Nn) return;
  int r = t / Nn, k = t % Nn;
  float s = b[k];
  for (int j = 0; j < K; ++j) s += A[(size_t)r * K + j] * W[(size_t)j * Nn + k];
  if (relu) s = fmaxf(s, 0.f);
  out[t] = s;
}

// ---------------------------------------------------------------------------
extern "C" void kernel_launch(void* const* d_in, const int* in_sizes, int n_in,
                              void* d_out, int out_size, void* d_ws,
                              size_t ws_size, hipStream_t stream) {
  const float* node_features = (const float*)d_in[0];
  const int*   node_types    = (const int*)d_in[1];
  const int*   edge_index    = (const int*)d_in[2];
  const float* edge_attr     = (const float*)d_in[3];
  const int*   comm_mask     = (const int*)d_in[4];
  const float* type_emb      = (const float*)d_in[5];
  const float* W0  = (const float*)d_in[6];
  const float* We0 = (const float*)d_in[7];
  const float* As0 = (const float*)d_in[8];
  const float* Ad0 = (const float*)d_in[9];
  const float* Ae0 = (const float*)d_in[10];
  const float* b0  = (const float*)d_in[11];
  const float* W1  = (const float*)d_in[12];
  const float* We1 = (const float*)d_in[13];
  const float* As1 = (const float*)d_in[14];
  const float* Ad1 = (const float*)d_in[15];
  const float* Ae1 = (const float*)d_in[16];
  const float* b1  = (const float*)d_in[17];
  const float* edge_gates = (const float*)d_in[18];
  const float* edge_attn  = (const float*)d_in[19];
  const float* pool_w = (const float*)d_in[20];
  const float* pool_b = (const float*)d_in[21];
  const float* mw1 = (const float*)d_in[22];
  const float* mb1 = (const float*)d_in[23];
  const float* mw2 = (const float*)d_in[24];
  const float* mb2 = (const float*)d_in[25];
  const float* mw3 = (const float*)d_in[26];
  const float* mb3 = (const float*)d_in[27];

  float* ws     = (float*)d_ws;
  float* xpad   = ws + OFF_XPAD;
  float* hlin   = ws + OFF_HLIN;
  float* oagg   = ws + OFF_OAGG;
  float* xc     = ws + OFF_XC;
  float* as_    = ws + OFF_AS;
  float* ad_    = ws + OFF_AD;
  float* den    = ws + OFF_DEN;
  float* ex     = ws + OFF_EX;
  float* sc     = ws + OFF_SC;
  float* misc   = ws + OFF_MISC;
  float* w0p    = ws + OFF_W0P;
  float* pooled = ws + OFF_POOL;
  float* z      = ws + OFF_Z;
  float* z1     = ws + OFF_Z1;
  float* z2     = ws + OFF_Z2;

  const int B = 256;
  auto cdiv = [](long long a, long long b) { return (int)((a + b - 1) / b); };
  const long long NH  = (long long)N_NODES * HID;
  const long long ET  = (long long)(E_EDGES + N_NODES);
  const long long ETH = ET * HID;

  scalars_kernel<<<1, 32, 0, stream>>>(edge_gates, edge_attn, misc);
  build_x_kernel<<<cdiv((long long)N_NODES * KPAD, B), B, 0, stream>>>(
      node_features, node_types, type_emb, xpad);
  zero_kernel<<<cdiv(NH, B), B, 0, stream>>>(xc, NH);

  for (int i = 0; i < 3; ++i) {
    // per-type scalar prep
    zero_kernel<<<1, 32, 0, stream>>>(misc + MISC_MSUM, EDIM);
    reduce_mean_kernel<<<cdiv(E_EDGES, B), B, 0, stream>>>(
        edge_attr + (size_t)i * E_EDGES * EDIM, misc + MISC_MSUM, E_EDGES);
    prep_type_kernel<<<1, 1, 0, stream>>>(We0, Ae0, We1, Ae1, misc, i,
                                          1.f / (float)E_EDGES);
    padw0_kernel<<<cdiv(KPAD * HID, B), B, 0, stream>>>(W0, w0p, i);

    // ---- layer 0 ----
    wmma_gemm_kernel<<<dim3(N_NODES / 16, HID / 16), 32, 0, stream>>>(
        xpad, w0p, hlin, KPAD, HID);
    alpha0_kernel<<<cdiv(N_NODES, B), B, 0, stream>>>(hlin, As0, Ad0, as_, ad_, i);
    zero_kernel<<<cdiv((long long)N_NODES * HHEADS, B), B, 0, stream>>>(
        den, (long long)N_NODES * HHEADS);
    zero_kernel<<<cdiv(NH, B), B, 0, stream>>>(oagg, NH);
    edgeA_kernel<4><<<cdiv(ET, B), B, 0, stream>>>(edge_index, edge_attr, misc,
                                                   as_, ad_, ex, den, i);
    edgeB_kernel<4><<<cdiv(ETH, B), B, 0, stream>>>(edge_index, hlin, ex, den,
                                                    oagg, i);
    elu_bias_kernel<<<cdiv(NH, B), B, 0, stream>>>(oagg, b0, i);

    // ---- layer 1 ----
    wmma_gemm_kernel<<<dim3(N_NODES / 16, HID / 16), 32, 0, stream>>>(
        oagg, W1 + (size_t)i * HID * HID, hlin, HID, HID);
    alpha1_kernel<<<cdiv(N_NODES, B), B, 0, stream>>>(hlin, As1, Ad1, as_, ad_, i);
    zero_kernel<<<cdiv(N_NODES, B), B, 0, stream>>>(den, N_NODES);
    zero_kernel<<<cdiv(NH, B), B, 0, stream>>>(oagg, NH);
    edgeA_kernel<1><<<cdiv(ET, B), B, 0, stream>>>(edge_index, edge_attr, misc,
                                                   as_, ad_, ex, den, i);
    edgeB_kernel<1><<<cdiv(ETH, B), B, 0, stream>>>(edge_index, hlin, ex, den,
                                                    oagg, i);
    accum_kernel<<<cdiv(NH, B), B, 0, stream>>>(xc, oagg, b1, misc, i);
  }

  // ---- pooling + MLP head ----
  scores_kernel<<<cdiv(N_NODES - R_RSU, B), B, 0, stream>>>(xc, pool_w, pool_b, sc);
  pool_kernel<<<R_RSU, 256, 0, stream>>>(xc, sc, comm_mask, pooled);
  buildz_kernel<<<cdiv(R_RSU * 2 * HID, B), B, 0, stream>>>(xc, pooled, z);
  small_mm_kernel<<<cdiv(R_RSU * HID, B), B, 0, stream>>>(z, mw1, mb1, z1,
                                                          R_RSU, 2 * HID, HID, 1);
  small_mm_kernel<<<cdiv(R_RSU * (HID / 2), B), B, 0, stream>>>(
      z1, mw2, mb2, z2, R_RSU, HID, HID / 2, 1);
  small_mm_kernel<<<cdiv(R_RSU * NACT, B), B, 0, stream>>>(
      z2, mw3, mb3, (float*)d_out, R_RSU, HID / 2, NACT, 0);
}